// TitansDualTowerSASRec_36060545417750
// MI455X (gfx1250) — compile-verified
//
#include <hip/hip_runtime.h>
#include <math.h>

typedef float v2f __attribute__((ext_vector_type(2)));
typedef float v8f __attribute__((ext_vector_type(8)));
typedef unsigned int u32x4 __attribute__((ext_vector_type(4)));
typedef int i32x4 __attribute__((ext_vector_type(4)));
typedef int i32x8 __attribute__((ext_vector_type(8)));

#define B_  32
#define T_  1024
#define D_  256
#define F_  1024
#define BT_ (B_ * T_)

#if defined(__gfx1250__) && __has_builtin(__builtin_amdgcn_wmma_f32_16x16x4_f32)
#define HAS_WMMA_F32X4 1
#else
#define HAS_WMMA_F32X4 0
#endif

#if defined(__gfx1250__) && __has_builtin(__builtin_amdgcn_tensor_load_to_lds)
#define HAS_TDM 1
#else
#define HAS_TDM 0
#endif

// ---------------------------------------------------------------------------
// TDM helper: DMA a 64x64 fp32 tile (row stride = strideElems) from global
// into LDS at byte offset ldsOff, inserting 4 pad DWORDs after every 64
// DWORDs so rows land at the 68-float anti-bank-conflict stride.
// D# layout per CDNA5 ISA ch.8 (group0: count/lds/global/type, group1:
// data_size=4B, pad_enable, pad_interval=64DW, pad_amount=4DW, tile 64x64).
// Toolchain here uses the 6-arg builtin (extra i32x8 group + cpol).
// ---------------------------------------------------------------------------
#if HAS_TDM
__device__ __forceinline__ void tdm_tile64(const float* gp, unsigned ldsOff,
                                           int strideElems) {
  const unsigned long long ga = (unsigned long long)(uintptr_t)gp;
  u32x4 g0;
  g0[0] = 1u;                                   // count=1, user mode
  g0[1] = ldsOff;                               // lds_addr (bytes)
  g0[2] = (unsigned)(ga & 0xFFFFFFFFu);         // global_addr[31:0]
  g0[3] = (unsigned)(ga >> 32) | 0x80000000u;   // global_addr[56:32] | type=2
  i32x8 g1;
  g1[0] = 0x00020000                            // data_size = 4B
        | (1 << 20)                             // pad_enable
        | (5 << 22)                             // pad_interval: 64 DWORDs
        | (3 << 25);                            // pad_amount: 4 DWORDs
  g1[1] = 0;                                    // tensor_dim0[15:0] (dim0=2^30)
  g1[2] = 0x4000;                               // tensor_dim0[31:16]
  g1[3] = 0x4000 | (64 << 16);                  // tensor_dim1=2^30; tile_dim0=64
  g1[4] = 64;                                   // tile_dim1=64, tile_dim2=0
  g1[5] = strideElems;                          // tensor_dim0_stride[31:0]
  g1[6] = 0;                                    // stride hi / dim1_stride lo
  g1[7] = 0;
  const i32x4 gz4 = {0, 0, 0, 0};               // groups 2/3 unused (2D)
  const i32x8 gz8 = {0, 0, 0, 0, 0, 0, 0, 0};   // extra group (6-arg form)
  __builtin_amdgcn_tensor_load_to_lds(g0, g1, gz4, gz4, gz8, 0);
}
#endif

template <int N>
__device__ __forceinline__ void wait_tensorcnt() {
#if defined(__gfx1250__)
#if __has_builtin(__builtin_amdgcn_s_wait_tensorcnt)
  __builtin_amdgcn_s_wait_tensorcnt(N);
#else
  asm volatile("s_wait_tensorcnt %0" ::"n"(N));
#endif
#endif
}

// ---------------------------------------------------------------------------
// LayerNorm: one wave per row of 256.
// ---------------------------------------------------------------------------
__global__ __launch_bounds__(256) void ln_kernel(const float* __restrict__ x,
                                                 const float* __restrict__ g,
                                                 const float* __restrict__ beta,
                                                 float* __restrict__ y) {
  const int row  = blockIdx.x * 8 + (threadIdx.x >> 5);
  const int lane = threadIdx.x & 31;
  const float* xr = x + (size_t)row * D_;
  float v[8];
  float s = 0.f, s2 = 0.f;
#pragma unroll
  for (int i = 0; i < 8; ++i) {
    v[i] = xr[lane + 32 * i];
    s += v[i];
    s2 += v[i] * v[i];
  }
#pragma unroll
  for (int m = 16; m; m >>= 1) {
    s  += __shfl_xor(s, m, 32);
    s2 += __shfl_xor(s2, m, 32);
  }
  const float mu  = s * (1.f / D_);
  const float var = s2 * (1.f / D_) - mu * mu;
  const float rs  = rsqrtf(var + 1e-5f);
  float* yr = y + (size_t)row * D_;
#pragma unroll
  for (int i = 0; i < 8; ++i) {
    const int j = lane + 32 * i;
    yr[j] = (v[i] - mu) * rs * g[j] + beta[j];
  }
}

// ---------------------------------------------------------------------------
// WMMA fp32 GEMM: C[m,n] = sum_k A[m,k]*W[n,k] (+epilogue).
// 64x64 tile, 4 waves. K chunks of 64 staged by TDM (double-buffered,
// DMA of chunk c+1 overlaps WMMA on chunk c) or by vector loads (fallback).
//   EPI 0: none | 1: relu(x+bias) | 2: x+bias+resid
// Dynamic LDS: 4 * 64*68 floats (2 A buffers + 2 W buffers).
// ---------------------------------------------------------------------------
#define GTILE (64 * 68)

template <int EPI>
__global__ __launch_bounds__(128) void gemm_wmma(const float* __restrict__ A,
                                                 const float* __restrict__ W,
                                                 const float* __restrict__ bias,
                                                 const float* __restrict__ resid,
                                                 float* __restrict__ C,
                                                 int M, int N, int K) {
  extern __shared__ float smem[];
  float* AsB0 = smem;
  float* AsB1 = smem + GTILE;
  float* WsB0 = smem + 2 * GTILE;
  float* WsB1 = smem + 3 * GTILE;

  const int m0   = blockIdx.x * 64;
  const int n0   = blockIdx.y * 64;
  const int tid  = threadIdx.x;
  const int wid  = tid >> 5;
  const int lane = tid & 31;
  const int lo16 = lane & 15;
  const int hi   = lane >> 4;

  v8f acc[4];
#pragma unroll
  for (int t = 0; t < 4; ++t)
#pragma unroll
    for (int e = 0; e < 8; ++e) acc[t][e] = 0.f;

  const int nc = K >> 6;

#if HAS_TDM
  if (wid == 0) {  // wave 0 drives the DMA engine
    tdm_tile64(&A[(size_t)m0 * K], (unsigned)(uintptr_t)AsB0, K);
    tdm_tile64(&W[(size_t)n0 * K], (unsigned)(uintptr_t)WsB0, K);
  }
#endif

  for (int c = 0; c < nc; ++c) {
    const int p = c & 1;
    float* Asp = p ? AsB1 : AsB0;
    float* Wsp = p ? WsB1 : WsB0;

#if HAS_TDM
    if (wid == 0) {
      if (c + 1 < nc) {  // prefetch next chunk into the other buffer
        const int kn = (c + 1) << 6;
        tdm_tile64(&A[(size_t)m0 * K + kn],
                   (unsigned)(uintptr_t)(p ? AsB0 : AsB1), K);
        tdm_tile64(&W[(size_t)n0 * K + kn],
                   (unsigned)(uintptr_t)(p ? WsB0 : WsB1), K);
        wait_tensorcnt<2>();  // chunk c's two tiles have landed
      } else {
        wait_tensorcnt<0>();
      }
    }
    __syncthreads();
#else
    const int kc = c << 6;
#pragma unroll
    for (int i = 0; i < 8; ++i) {
      const int idx = tid + 128 * i;
      const int row = idx >> 4;
      const int q4  = idx & 15;
      const float4 av = *(const float4*)&A[(size_t)(m0 + row) * K + kc + q4 * 4];
      *(float4*)&Asp[row * 68 + q4 * 4] = av;
      const float4 bv = *(const float4*)&W[(size_t)(n0 + row) * K + kc + q4 * 4];
      *(float4*)&Wsp[row * 68 + q4 * 4] = bv;
    }
    __syncthreads();
#endif

#pragma unroll
    for (int kk = 0; kk < 64; kk += 4) {
      const int krow = kk + 2 * hi;
      const v2f a = *(const v2f*)&Asp[(16 * wid + lo16) * 68 + krow];
#pragma unroll
      for (int t = 0; t < 4; ++t) {
        const v2f b = *(const v2f*)&Wsp[(16 * t + lo16) * 68 + krow];
#if HAS_WMMA_F32X4
        acc[t] = __builtin_amdgcn_wmma_f32_16x16x4_f32(
            false, a, false, b, (short)0, acc[t], false, false);
#else
        acc[t][0] += a[0] * b[0];
#endif
      }
    }
    __syncthreads();  // everyone done reading buffer p before DMA reuses it
  }

#pragma unroll
  for (int t = 0; t < 4; ++t) {
    const int n = n0 + 16 * t + lo16;
#pragma unroll
    for (int r = 0; r < 8; ++r) {
      const int m = m0 + 16 * wid + r + 8 * hi;
      float val = acc[t][r];
      if (EPI == 1) val = fmaxf(val + bias[n], 0.f);
      if (EPI == 2) val = val + bias[n] + resid[(size_t)m * N + n];
      C[(size_t)m * N + n] = val;
    }
  }
}

// ---------------------------------------------------------------------------
// Row ops: l2-normalize q,k rows in place; alpha/eta gates. One wave per row.
// ---------------------------------------------------------------------------
__global__ __launch_bounds__(256) void rowops_kernel(
    const float* __restrict__ xn, float* __restrict__ q, float* __restrict__ k,
    const float* __restrict__ Wa, const float* __restrict__ ba,
    const float* __restrict__ We, const float* __restrict__ be,
    float* __restrict__ alpha, float* __restrict__ eta) {
  const int row  = blockIdx.x * 8 + (threadIdx.x >> 5);
  const int lane = threadIdx.x & 31;
  const float* xr = xn + (size_t)row * D_;
  float* qr = q + (size_t)row * D_;
  float* kr = k + (size_t)row * D_;
  float qv[8], kv[8];
  float sq = 0.f, sk = 0.f, da = 0.f, de = 0.f;
#pragma unroll
  for (int i = 0; i < 8; ++i) {
    const int j = lane + 32 * i;
    const float xv = xr[j];
    qv[i] = qr[j];
    kv[i] = kr[j];
    sq += qv[i] * qv[i];
    sk += kv[i] * kv[i];
    da += xv * Wa[j];
    de += xv * We[j];
  }
#pragma unroll
  for (int m = 16; m; m >>= 1) {
    sq += __shfl_xor(sq, m, 32);
    sk += __shfl_xor(sk, m, 32);
    da += __shfl_xor(da, m, 32);
    de += __shfl_xor(de, m, 32);
  }
  const float qscale = 1.f / fmaxf(sqrtf(sq), 1e-12f);
  const float kscale = 1.f / fmaxf(sqrtf(sk), 1e-12f);
#pragma unroll
  for (int i = 0; i < 8; ++i) {
    const int j = lane + 32 * i;
    qr[j] = qv[i] * qscale;
    kr[j] = kv[i] * kscale;
  }
  if (lane == 0) {
    alpha[row] = 1.f / (1.f + __expf(-(da + ba[0])));
    eta[row]   = 1.f / (1.f + __expf(-(de + be[0])));
  }
}

// ---------------------------------------------------------------------------
// Delta-rule scan: one block per batch, state [256,256] register-resident as
// float2 so the hot loops can select v_pk_fma_f32 (packed fp32, 2x VALU rate).
// 512 threads: thread (ty=tid>>4, tx=tid&15) owns an 8x16 block of S.
// ---------------------------------------------------------------------------
#define RP_STRIDE 260

__global__ __launch_bounds__(512) void recur_kernel(
    const float* __restrict__ q, const float* __restrict__ k,
    const float* __restrict__ v, const float* __restrict__ alpha,
    const float* __restrict__ eta, float* __restrict__ reads) {
  extern __shared__ float sh[];
  float* rowpart = sh;                        // [16][RP_STRIDE] pred partials
  float* colpart = rowpart + 16 * RP_STRIDE;  // [32][RP_STRIDE] read partials
  float* qs   = colpart + 32 * RP_STRIDE;     // 256
  float* ks   = qs + 256;                     // 256
  float* vs   = ks + 256;                     // 256
  float* errv = vs + 256;                     // 256

  const int b   = blockIdx.x;
  const int tid = threadIdx.x;
  const int ty  = tid >> 4;  // rows 8*ty..+7
  const int tx  = tid & 15;  // cols 16*tx..+15
  const int r0  = ty * 8;
  const int c0  = tx * 16;

  v2f S2[8][8];
#pragma unroll
  for (int i = 0; i < 8; ++i)
#pragma unroll
    for (int j = 0; j < 8; ++j) S2[i][j] = (v2f){0.f, 0.f};

  const size_t base = (size_t)b * T_ * D_;
  for (int t = 0; t < T_; ++t) {
    const size_t rb = base + (size_t)t * D_;
    if (tid < 256) {
      qs[tid] = q[rb + tid];
      ks[tid] = k[rb + tid];
      vs[tid] = v[rb + tid];
      if (t + 1 < T_) {
        __builtin_prefetch(&q[rb + D_ + tid], 0, 0);
        __builtin_prefetch(&k[rb + D_ + tid], 0, 0);
        __builtin_prefetch(&v[rb + D_ + tid], 0, 0);
      }
    }
    __syncthreads();

    float qv[8];
    v2f kv2[8];
#pragma unroll
    for (int i = 0; i < 8; ++i) qv[i] = qs[r0 + i];
#pragma unroll
    for (int j = 0; j < 8; ++j) kv2[j] = *(const v2f*)&ks[c0 + 2 * j];

    v2f cr2[8];
    float pr[8];
#pragma unroll
    for (int j = 0; j < 8; ++j) cr2[j] = (v2f){0.f, 0.f};
#pragma unroll
    for (int i = 0; i < 8; ++i) {
      v2f pa = {0.f, 0.f};
      const v2f qq = {qv[i], qv[i]};
#pragma unroll
      for (int j = 0; j < 8; ++j) {
        pa = __builtin_elementwise_fma(S2[i][j], kv2[j], pa);       // (S k)_i
        cr2[j] = __builtin_elementwise_fma(qq, S2[i][j], cr2[j]);   // (S^T q)_j
      }
      pr[i] = pa[0] + pa[1];
    }
#pragma unroll
    for (int i = 0; i < 8; ++i) rowpart[tx * RP_STRIDE + r0 + i] = pr[i];
#pragma unroll
    for (int j = 0; j < 8; ++j)
      *(v2f*)&colpart[ty * RP_STRIDE + c0 + 2 * j] = cr2[j];
    __syncthreads();

    if (tid < 256) {
      float s = 0.f;
#pragma unroll
      for (int p = 0; p < 32; ++p) s += colpart[p * RP_STRIDE + tid];
      reads[rb + tid] = s;  // read_t from pre-update state
    } else {
      const int i = tid - 256;
      float s = 0.f;
#pragma unroll
      for (int p = 0; p < 16; ++p) s += rowpart[p * RP_STRIDE + i];
      errv[i] = vs[i] - s;  // v - S k
    }
    __syncthreads();

    const float a   = alpha[b * T_ + t];
    const float e   = eta[b * T_ + t];
    const float oma = 1.f - a;
    const v2f oma2  = {oma, oma};
    float ee[8];
#pragma unroll
    for (int i = 0; i < 8; ++i) ee[i] = e * errv[r0 + i];
#pragma unroll
    for (int i = 0; i < 8; ++i) {
      const v2f ee2 = {ee[i], ee[i]};
#pragma unroll
      for (int j = 0; j < 8; ++j)
        S2[i][j] = __builtin_elementwise_fma(ee2, kv2[j], oma2 * S2[i][j]);
    }
    __syncthreads();
  }
}

// ---------------------------------------------------------------------------
extern "C" void kernel_launch(void* const* d_in, const int* in_sizes, int n_in,
                              void* d_out, int out_size, void* d_ws, size_t ws_size,
                              hipStream_t stream) {
  const float* x     = (const float*)d_in[0];
  const float* Wq    = (const float*)d_in[1];
  const float* Wk    = (const float*)d_in[2];
  const float* Wv    = (const float*)d_in[3];
  const float* Wa    = (const float*)d_in[4];
  const float* ba    = (const float*)d_in[5];
  const float* We    = (const float*)d_in[6];
  const float* be    = (const float*)d_in[7];
  const float* Wo    = (const float*)d_in[8];
  const float* bo    = (const float*)d_in[9];
  const float* W1    = (const float*)d_in[10];
  const float* b1    = (const float*)d_in[11];
  const float* W2    = (const float*)d_in[12];
  const float* b2    = (const float*)d_in[13];
  const float* g1    = (const float*)d_in[14];
  const float* beta1 = (const float*)d_in[15];
  const float* g2    = (const float*)d_in[16];
  const float* beta2 = (const float*)d_in[17];
  float* out = (float*)d_out;

  float* ws    = (float*)d_ws;
  float* xn    = ws;                     // [BT,D]
  float* qb    = xn + (size_t)BT_ * D_;  // [BT,D]
  float* kb    = qb + (size_t)BT_ * D_;  // [BT,D]
  float* vb    = kb + (size_t)BT_ * D_;  // [BT,D]
  float* alpha = vb + (size_t)BT_ * D_;  // [BT]
  float* eta   = alpha + BT_;            // [BT]
  float* hbuf  = eta + BT_;              // [BT,F]
  float* reads = xn;  // xn dead after rowops
  float* xo    = qb;  // qb dead after scan
  float* x2    = kb;  // kb dead after scan

  const dim3 blkLN(256), blkG(128), blkR(512);
  const size_t gemmLds = (size_t)4 * GTILE * sizeof(float);  // 69632 B
  const size_t recLds  = (size_t)(48 * RP_STRIDE + 4 * 256) * sizeof(float);

  ln_kernel<<<dim3(BT_ / 8), blkLN, 0, stream>>>(x, g1, beta1, xn);
  gemm_wmma<0><<<dim3(BT_ / 64, D_ / 64), blkG, gemmLds, stream>>>(xn, Wq, nullptr, nullptr, qb, BT_, D_, D_);
  gemm_wmma<0><<<dim3(BT_ / 64, D_ / 64), blkG, gemmLds, stream>>>(xn, Wk, nullptr, nullptr, kb, BT_, D_, D_);
  gemm_wmma<0><<<dim3(BT_ / 64, D_ / 64), blkG, gemmLds, stream>>>(xn, Wv, nullptr, nullptr, vb, BT_, D_, D_);
  rowops_kernel<<<dim3(BT_ / 8), blkLN, 0, stream>>>(xn, qb, kb, Wa, ba, We, be, alpha, eta);
  recur_kernel<<<dim3(B_), blkR, recLds, stream>>>(qb, kb, vb, alpha, eta, reads);
  gemm_wmma<2><<<dim3(BT_ / 64, D_ / 64), blkG, gemmLds, stream>>>(reads, Wo, bo, x, xo, BT_, D_, D_);
  ln_kernel<<<dim3(BT_ / 8), blkLN, 0, stream>>>(xo, g2, beta2, x2);
  gemm_wmma<1><<<dim3(BT_ / 64, F_ / 64), blkG, gemmLds, stream>>>(x2, W1, b1, nullptr, hbuf, BT_, F_, D_);
  gemm_wmma<2><<<dim3(BT_ / 64, D_ / 64), blkG, gemmLds, stream>>>(hbuf, W2, b2, xo, out, BT_, D_, F_);
}